// GraphUpdate_48893907697803
// MI455X (gfx1250) — compile-verified
//
#include <hip/hip_runtime.h>
#include <hip/hip_bf16.h>

typedef float v2f __attribute__((ext_vector_type(2)));
typedef float v8f __attribute__((ext_vector_type(8)));

// ---------------------------------------------------------------------------
// Small utility kernels
// ---------------------------------------------------------------------------
__global__ void k_fill_i32(int* __restrict__ p, int v, int n) {
    int t = blockIdx.x * blockDim.x + threadIdx.x;
    if (t < n) p[t] = v;
}

__global__ void k_zero_f32(float* __restrict__ p, long long n) {
    long long t = (long long)blockIdx.x * blockDim.x + threadIdx.x;
    if (t < n) p[t] = 0.0f;
}

__global__ void k_count_deg(const int* __restrict__ dst, int* __restrict__ deg, int E) {
    int t = blockIdx.x * blockDim.x + threadIdx.x;
    if (t < E) atomicAdd(&deg[dst[t]], 1);
}

__global__ void k_dinv(const int* __restrict__ deg, float* __restrict__ dinv, int n) {
    int t = blockIdx.x * blockDim.x + threadIdx.x;
    if (t < n) {
        float d = (float)deg[t];            // always >= 1 (self loop)
        dinv[t] = rsqrtf(d);
    }
}

// ---------------------------------------------------------------------------
// FP32 WMMA GEMM:  D[M,N] = A[M,K] * B[K,N], row-major, one wave per 16x16 tile.
// Uses V_WMMA_F32_16X16X4_F32. M, N, K multiples of 16/16/4 (true here:
// 50000 x 64 x 128 and 50000 x 128 x 64).
// VGPR layouts per CDNA5 ISA 7.12.2:
//   A 16x4 : lane m=lane&15; lanes<16 hold K=k0+0/1, lanes>=16 hold K=k0+2/3
//   B 4x16 : lane n=lane&15; lanes<16 hold K rows k0+0/1, lanes>=16 rows k0+2/3
//   C/D    : VGPR i, lanes<16 -> (M=i, N=lane), lanes>=16 -> (M=i+8, N=lane-16)
// ---------------------------------------------------------------------------
__global__ void k_gemm_wmma(const float* __restrict__ A, const float* __restrict__ B,
                            float* __restrict__ D, int M, int K, int N) {
    const int lane   = threadIdx.x & 31;
    const int waveId = threadIdx.x >> 5;
    const int wavesPerBlock = blockDim.x >> 5;
    const int tilesN = N >> 4;
    const int totalTiles = (M >> 4) * tilesN;

    int tile = blockIdx.x * wavesPerBlock + waveId;   // wave-uniform
    if (tile >= totalTiles) return;                   // wave-uniform branch

    const int mt = tile / tilesN;
    const int nt = tile - mt * tilesN;

    const int mn  = lane & 15;                // row (for A) / col (for B)
    const int kb  = (lane >> 4) << 1;         // 0 for lanes 0-15, 2 for lanes 16-31

    const float* arow = A + (long long)(mt * 16 + mn) * K;
    const float* bcol = B + (nt * 16 + mn);

    v8f acc = {};
    for (int k0 = 0; k0 < K; k0 += 4) {
        v2f a, b;
        a.x = arow[k0 + kb];
        a.y = arow[k0 + kb + 1];
        b.x = bcol[(long long)(k0 + kb) * N];
        b.y = bcol[(long long)(k0 + kb + 1) * N];
        acc = __builtin_amdgcn_wmma_f32_16x16x4_f32(
            /*neg_a=*/false, a, /*neg_b=*/false, b,
            /*c_mod=*/(short)0, acc, /*reuse_a=*/false, /*reuse_b=*/false);
    }

    const int nbase = nt * 16 + mn;
    const int mbase = mt * 16 + ((lane >> 4) << 3);
#pragma unroll
    for (int i = 0; i < 8; ++i)
        D[(long long)(mbase + i) * N + nbase] = acc[i];
}

// ---------------------------------------------------------------------------
// Edge scatter: for each edge e, agg[dst[e]] += H[src[e]] * dinv[src]*dinv[dst]
// One thread handles one float4 chunk of one edge. chunkShift = log2(C/4):
//   C=128 -> 32 chunks (full wave per edge), C=64 -> 16 chunks.
// ---------------------------------------------------------------------------
__global__ void k_scatter(const float* __restrict__ H, const float* __restrict__ dinv,
                          const int* __restrict__ src, const int* __restrict__ dst,
                          float* __restrict__ agg, int E, int C, int chunkShift) {
    long long t = (long long)blockIdx.x * blockDim.x + threadIdx.x;
    long long total = (long long)E << chunkShift;
    if (t >= total) return;
    int e  = (int)(t >> chunkShift);
    int c4 = (int)(t & ((1 << chunkShift) - 1));
    int s = src[e];
    int d = dst[e];
    float norm = dinv[s] * dinv[d];
    const float4 v = ((const float4*)(H + (long long)s * C))[c4];
    float* base = agg + (long long)d * C + (c4 << 2);
    unsafeAtomicAdd(base + 0, v.x * norm);   // global_atomic_add_f32
    unsafeAtomicAdd(base + 1, v.y * norm);
    unsafeAtomicAdd(base + 2, v.z * norm);
    unsafeAtomicAdd(base + 3, v.w * norm);
}

// ---------------------------------------------------------------------------
// Finalize: agg += self-loop term + bias, optional ReLU (in place).
// cShift = log2(C).
// ---------------------------------------------------------------------------
__global__ void k_finalize(float* __restrict__ agg, const float* __restrict__ H,
                           const float* __restrict__ dinv, const float* __restrict__ bias,
                           long long total, int C, int cShift, int doRelu) {
    long long t = (long long)blockIdx.x * blockDim.x + threadIdx.x;
    if (t >= total) return;
    int i = (int)(t >> cShift);
    int c = (int)(t & (C - 1));
    float di = dinv[i];
    float v = agg[t] + H[t] * (di * di) + bias[c];
    if (doRelu) v = fmaxf(v, 0.0f);
    agg[t] = v;
}

// ---------------------------------------------------------------------------
// Launcher
// ---------------------------------------------------------------------------
extern "C" void kernel_launch(void* const* d_in, const int* in_sizes, int n_in,
                              void* d_out, int out_size, void* d_ws, size_t ws_size,
                              hipStream_t stream) {
    const float* x  = (const float*)d_in[0];
    const int*   ei = (const int*)  d_in[1];
    const float* W1 = (const float*)d_in[2];
    const float* b1 = (const float*)d_in[3];
    const float* W2 = (const float*)d_in[4];
    const float* b2 = (const float*)d_in[5];
    float* out = (float*)d_out;

    const int HID  = in_sizes[3];            // 128
    const int OUTC = in_sizes[5];            // 64
    const int INC  = in_sizes[2] / HID;      // 64
    const int N    = in_sizes[0] / INC;      // 50000
    const int E    = in_sizes[1] / 2;        // 1,600,000
    const int* src = ei;
    const int* dst = ei + E;

    // workspace layout (256B aligned regions)
    auto alignup = [](size_t v) { return (v + 255) & ~(size_t)255; };
    size_t o = 0;
    int*   deg  = (int*)  ((char*)d_ws + o); o = alignup(o + (size_t)N * 4);
    float* dinv = (float*)((char*)d_ws + o); o = alignup(o + (size_t)N * 4);
    float* h1   = (float*)((char*)d_ws + o); o = alignup(o + (size_t)N * HID * 4);
    float* agg1 = (float*)((char*)d_ws + o); o = alignup(o + (size_t)N * HID * 4);
    float* h2   = h1;                        // reuse h1 region for layer-2 GEMM output

    const int TB = 256;
    auto blocks = [&](long long n) { return (int)((n + TB - 1) / TB); };

    // --- degree + normalization (deg starts at 1 for the self loop) ---
    k_fill_i32 <<<blocks(N), TB, 0, stream>>>(deg, 1, N);
    k_count_deg<<<blocks(E), TB, 0, stream>>>(dst, deg, E);
    k_dinv     <<<blocks(N), TB, 0, stream>>>(deg, dinv, N);

    // --- layer 1: h1 = x @ W1 ---
    {
        int tiles = (N >> 4) * (HID >> 4);          // 25000
        int wavesPerBlock = TB >> 5;
        k_gemm_wmma<<<(tiles + wavesPerBlock - 1) / wavesPerBlock, TB, 0, stream>>>(
            x, W1, h1, N, INC, HID);
    }
    long long n1 = (long long)N * HID;
    k_zero_f32<<<blocks(n1), TB, 0, stream>>>(agg1, n1);
    k_scatter <<<blocks((long long)E << 5), TB, 0, stream>>>(h1, dinv, src, dst, agg1,
                                                             E, HID, /*chunkShift=*/5);
    k_finalize<<<blocks(n1), TB, 0, stream>>>(agg1, h1, dinv, b1, n1, HID, /*cShift=*/7,
                                              /*doRelu=*/1);

    // --- layer 2: h2 = relu(h) @ W2 ---
    {
        int tiles = (N >> 4) * (OUTC >> 4);         // 12500
        int wavesPerBlock = TB >> 5;
        k_gemm_wmma<<<(tiles + wavesPerBlock - 1) / wavesPerBlock, TB, 0, stream>>>(
            agg1, W2, h2, N, HID, OUTC);
    }
    long long n2 = (long long)N * OUTC;
    k_zero_f32<<<blocks(n2), TB, 0, stream>>>(out, n2);
    k_scatter <<<blocks((long long)E << 4), TB, 0, stream>>>(h2, dinv, src, dst, out,
                                                             E, OUTC, /*chunkShift=*/4);
    k_finalize<<<blocks(n2), TB, 0, stream>>>(out, h2, dinv, b2, n2, OUTC, /*cShift=*/6,
                                              /*doRelu=*/0);
}